// Dense_369367187664
// MI455X (gfx1250) — compile-verified
//
#include <hip/hip_runtime.h>
#include <math.h>

// Problem constants (match reference)
constexpr int cB = 8;
constexpr int cS = 2048;
constexpr int cH = 1024;
constexpr int cROWS = cB * cS;         // 16384 query rows
constexpr float cRSQRT_H = 0.03125f;   // 1/sqrt(1024)

typedef __attribute__((ext_vector_type(2))) float v2f;
typedef __attribute__((ext_vector_type(8))) float v8f;

// ---------------------------------------------------------------------------
// Kernel 1: wqk[h] = sum_o Wq[o,h] * Wk[o]   (fused projection vector, 4 MB read)
// ---------------------------------------------------------------------------
__global__ void k_wqk(const float* __restrict__ Wq, const float* __restrict__ Wk,
                      float* __restrict__ wqk) {
  int h = blockIdx.x * blockDim.x + threadIdx.x;
  float a0 = 0.f, a1 = 0.f, a2 = 0.f, a3 = 0.f;  // break FMA dependence chain
  for (int o = 0; o < cH; o += 4) {
    a0 += Wq[(size_t)(o + 0) * cH + h] * Wk[o + 0];
    a1 += Wq[(size_t)(o + 1) * cH + h] * Wk[o + 1];
    a2 += Wq[(size_t)(o + 2) * cH + h] * Wk[o + 2];
    a3 += Wq[(size_t)(o + 3) * cH + h] * Wk[o + 3];
  }
  wqk[h] = (a0 + a1) + (a2 + a3);
}

// ---------------------------------------------------------------------------
// Kernel 2: block 0 -> bqk = dot(bq, Wk); blocks 1..B -> per-batch min/max of sqi
// ---------------------------------------------------------------------------
__global__ void k_scalars(const float* __restrict__ bq, const float* __restrict__ Wk,
                          const float* __restrict__ sqi,
                          float* __restrict__ bqk,
                          float* __restrict__ smin, float* __restrict__ smax) {
  __shared__ float red[256];
  __shared__ float rmn[256];
  __shared__ float rmx[256];
  int t = threadIdx.x;
  if (blockIdx.x == 0) {
    float acc = 0.f;
    for (int o = t; o < cH; o += 256) acc += bq[o] * Wk[o];
    red[t] = acc;
    __syncthreads();
    for (int s = 128; s > 0; s >>= 1) {
      if (t < s) red[t] += red[t + s];
      __syncthreads();
    }
    if (t == 0) bqk[0] = red[0];
  } else {
    int b = blockIdx.x - 1;
    const float* p = sqi + (size_t)b * cS;
    float mn = 3.4e38f, mx = -3.4e38f;
    for (int j = t; j < cS; j += 256) {
      float v = p[j];
      mn = fminf(mn, v);
      mx = fmaxf(mx, v);
    }
    rmn[t] = mn; rmx[t] = mx;
    __syncthreads();
    for (int s = 128; s > 0; s >>= 1) {
      if (t < s) { rmn[t] = fminf(rmn[t], rmn[t + s]); rmx[t] = fmaxf(rmx[t], rmx[t + s]); }
      __syncthreads();
    }
    if (t == 0) { smin[b] = rmn[0]; smax[b] = rmx[0]; }
  }
}

// ---------------------------------------------------------------------------
// Kernel 3: a[r] = (dot(x_r, wqk) + bqk) / sqrt(H)    (one wave32 per row)
//           x_r = hidden[b,i,:] + pos_enc[i,:]
// ---------------------------------------------------------------------------
__global__ void k_arow(const float* __restrict__ hidden, const float* __restrict__ pos,
                       const float* __restrict__ wqk, const float* __restrict__ bqkp,
                       float* __restrict__ a) {
  int wave = threadIdx.x >> 5;
  int lane = threadIdx.x & 31;
  int r = blockIdx.x * 8 + wave;          // global row, r = b*S + i
  int i = r & (cS - 1);
  const float4* h4 = reinterpret_cast<const float4*>(hidden + (size_t)r * cH);
  const float4* p4 = reinterpret_cast<const float4*>(pos + (size_t)i * cH);
  const float4* w4 = reinterpret_cast<const float4*>(wqk);
  float acc = 0.f;
  for (int t = lane; t < cH / 4; t += 32) {
    float4 hv = h4[t], pv = p4[t], wv = w4[t];
    acc += (hv.x + pv.x) * wv.x + (hv.y + pv.y) * wv.y +
           (hv.z + pv.z) * wv.z + (hv.w + pv.w) * wv.w;
  }
  for (int off = 16; off > 0; off >>= 1) acc += __shfl_xor(acc, off, 32);
  if (lane == 0) a[r] = (acc + bqkp[0]) * cRSQRT_H;
}

// ---------------------------------------------------------------------------
// Kernel 4: per query row: attn[b,i,j] = softmax_j(a_i * sqi[b,j])
//           Nontemporal-store attn (write-once, 134 MB) and accumulate
//           w[b,j] += attn[b,i,j]/S  (column means of attn).
//           One wave per row, 8 rows per block (all same b); sqi cached in LDS.
// ---------------------------------------------------------------------------
__global__ void k_softmax(const float* __restrict__ a, const float* __restrict__ sqi,
                          const float* __restrict__ smin, const float* __restrict__ smax,
                          float* __restrict__ attn, float* __restrict__ w) {
  __shared__ float sq[cS];   // 8 KB
  __shared__ float wp[cS];   // 8 KB (per-block partial column sums)
  int b = blockIdx.x >> 8;           // 256 blocks per batch
  int i0 = (blockIdx.x & 255) * 8;
  int t = threadIdx.x;
  const float* sp = sqi + (size_t)b * cS;
  for (int j = t; j < cS; j += 256) { sq[j] = sp[j]; wp[j] = 0.f; }
  __syncthreads();

  int wave = t >> 5, lane = t & 31;
  int i = i0 + wave;
  float ai = a[(size_t)b * cS + i];
  // max_j(ai*sqi_j) without a scan: sign-select against precomputed extrema
  float m = (ai >= 0.f) ? ai * smax[b] : ai * smin[b];

  float sum = 0.f;
  for (int j = lane; j < cS; j += 32) sum += __expf(ai * sq[j] - m);
  for (int off = 16; off > 0; off >>= 1) sum += __shfl_xor(sum, off, 32);
  float inv = 1.0f / sum;

  float* arow = attn + ((size_t)(b * cS + i)) * cS;
  const float invS = 1.0f / (float)cS;
  for (int j = lane; j < cS; j += 32) {
    float e = __expf(ai * sq[j] - m) * inv;
    __builtin_nontemporal_store(e, arow + j);
    atomicAdd(&wp[j], e * invS);
  }
  __syncthreads();
  float* wb = w + (size_t)b * cS;
  for (int j = t; j < cS; j += 256) atomicAdd(&wb[j], wp[j]);
}

// ---------------------------------------------------------------------------
// Kernel 5: y[b,h] = sum_j w[b,j] * (hidden[b,j,h] + pos[j,h])
//           j split into chunks for parallelism; partials via global atomics.
// ---------------------------------------------------------------------------
constexpr int cJC = 8;   // j-chunks
__global__ void k_y(const float* __restrict__ hidden, const float* __restrict__ pos,
                    const float* __restrict__ w, float* __restrict__ yp) {
  __shared__ float ws_[cS / cJC];
  int bi = blockIdx.x;
  int jc = bi % cJC;            bi /= cJC;
  int hc = bi % (cH / 256);     int b = bi / (cH / 256);
  int h = hc * 256 + threadIdx.x;
  int j0 = jc * (cS / cJC);
  for (int j = threadIdx.x; j < cS / cJC; j += 256) ws_[j] = w[(size_t)b * cS + j0 + j];
  __syncthreads();
  const float* hb = hidden + ((size_t)b * cS + j0) * cH + h;
  const float* pb = pos + (size_t)j0 * cH + h;
  float acc = 0.f;
  for (int j = 0; j < cS / cJC; ++j)
    acc += ws_[j] * (hb[(size_t)j * cH] + pb[(size_t)j * cH]);
  atomicAdd(&yp[(size_t)b * cH + h], acc);
}

// ---------------------------------------------------------------------------
// Kernel 6: context = y(16x1024, rows 8..15 zero) * Wv^T + bv
//           via V_WMMA_F32_16X16X4_F32: one wave per 16x16 output tile,
//           K-loop of 256 WMMA ops. Fragment layouts per ISA 7.12.2.
// ---------------------------------------------------------------------------
__global__ void k_ctx_wmma(const float* __restrict__ yp, const float* __restrict__ Wv,
                           const float* __restrict__ bv, float* __restrict__ out) {
  int n0 = blockIdx.x * 16;        // output column tile (o dimension)
  int lane = threadIdx.x & 31;
  int mn = lane & 15;              // A: row M; B: col N
  int kb = (lane >> 4) * 2;        // lane-half selects K pair {0,1} or {2,3}
  v8f c = {0.f, 0.f, 0.f, 0.f, 0.f, 0.f, 0.f, 0.f};
  for (int k0 = 0; k0 < cH; k0 += 4) {
    v2f av, bvv;
    av[0] = yp[(size_t)mn * cH + k0 + kb];
    av[1] = yp[(size_t)mn * cH + k0 + kb + 1];
    bvv[0] = Wv[(size_t)(n0 + mn) * cH + k0 + kb];      // B[k][n] = Wv[n,k]
    bvv[1] = Wv[(size_t)(n0 + mn) * cH + k0 + kb + 1];
    c = __builtin_amdgcn_wmma_f32_16x16x4_f32(
        /*neg_a=*/false, av, /*neg_b=*/false, bvv,
        /*c_mod=*/(short)0, c, /*reuse_a=*/false, /*reuse_b=*/false);
  }
  // C/D layout: VGPR r -> M = r + 8*(lane>=16); N = n0 + (lane&15)
  int n = n0 + mn;
  int mbase = (lane >> 4) * 8;
  float bias = bv[n];
#pragma unroll
  for (int r = 0; r < 8; ++r) {
    int mm = mbase + r;
    if (mm < cB) out[(size_t)mm * cH + n] = c[r] + bias;
  }
}

// ---------------------------------------------------------------------------
extern "C" void kernel_launch(void* const* d_in, const int* in_sizes, int n_in,
                              void* d_out, int out_size, void* d_ws, size_t ws_size,
                              hipStream_t stream) {
  const float* hidden = (const float*)d_in[0];   // [B,S,H]
  const float* sqi    = (const float*)d_in[1];   // [B,1,S]
  const float* pos    = (const float*)d_in[2];   // [1,S,H]
  const float* Wq     = (const float*)d_in[3];   // [H,H]
  const float* bq     = (const float*)d_in[4];   // [H]
  const float* Wk     = (const float*)d_in[5];   // [H,1]
  // d_in[6] = bk: provably unused (uniform shift, cancels in softmax)
  const float* Wv     = (const float*)d_in[7];   // [H,H]
  const float* bv     = (const float*)d_in[8];   // [H]

  float* out_ctx  = (float*)d_out;                          // [B,H]
  float* out_attn = (float*)d_out + (size_t)cB * cH;        // [B,S,S]

  // Workspace layout (floats)
  float* ws   = (float*)d_ws;
  float* wqk  = ws;                         // H
  float* bqk  = ws + cH;                    // 1
  float* smin = ws + cH + 64;               // B
  float* smax = ws + cH + 128;              // B
  float* a    = ws + 2048;                  // B*S
  float* w    = a + cROWS;                  // B*S  (attn column means)
  float* yp   = w + cROWS;                  // 16*H (padded y, rows 8..15 zero)

  hipMemsetAsync(w,  0, (size_t)cROWS * sizeof(float), stream);
  hipMemsetAsync(yp, 0, (size_t)16 * cH * sizeof(float), stream);

  k_wqk<<<cH / 256, 256, 0, stream>>>(Wq, Wk, wqk);
  k_scalars<<<1 + cB, 256, 0, stream>>>(bq, Wk, sqi, bqk, smin, smax);
  k_arow<<<cROWS / 8, 256, 0, stream>>>(hidden, pos, wqk, bqk, a);
  k_softmax<<<cB * (cS / 8), 256, 0, stream>>>(a, sqi, smin, smax, out_attn, w);
  k_y<<<cB * (cH / 256) * cJC, 256, 0, stream>>>(hidden, pos, w, yp);
  k_ctx_wmma<<<cH / 16, 32, 0, stream>>>(yp, Wv, bv, out_ctx);
}